// CrystalRGCNVAE_79637283602841
// MI455X (gfx1250) — compile-verified
//
#include <hip/hip_runtime.h>
#include <math.h>

// ---------------- static problem configuration ----------------
#define NB    256          // graphs
#define NPER  128          // nodes per graph
#define NN    (NB*NPER)    // 32768 nodes
#define EPER  2048         // edges per graph
#define EE    (NB*EPER)    // 524288 edges
#define HID   64
#define LAT   32

typedef float v2f __attribute__((ext_vector_type(2)));
typedef float v8f __attribute__((ext_vector_type(8)));

// f32 WMMA: D(16x16) = A(16x4) * B(4x16) + C   (CDNA5 V_WMMA_F32_16X16X4_F32)
__device__ inline v8f wmma4(v2f a, v2f b, v8f c) {
  return __builtin_amdgcn_wmma_f32_16x16x4_f32(
      /*neg_a=*/false, a, /*neg_b=*/false, b,
      /*c_mod=*/(short)0, c, /*reuse_a=*/false, /*reuse_b=*/false);
}

__device__ inline float lrelu(float x) { return x > 0.f ? x : 0.1f * x; }
__device__ inline float relu_(float x) { return x > 0.f ? x : 0.f; }

// ---------------- K0: fold basis decomposition into dense weights ----------------
// W1cat [20][64]  = rows r*4+i : sum_b comp1[r,b]*basis1[b,i,:], rows 16..19: loop1
// W2cat [320][128]= rows r*64+i: sum_b comp2[r,b]*basis2[b,i,:], rows 256..319: loop2
__global__ void prep_w_kernel(const float* __restrict__ basis1, const float* __restrict__ comp1,
                              const float* __restrict__ loop1,
                              const float* __restrict__ basis2, const float* __restrict__ comp2,
                              const float* __restrict__ loop2,
                              float* __restrict__ W1cat, float* __restrict__ W2cat) {
  int tid = blockIdx.x * blockDim.x + threadIdx.x;
  int stride = gridDim.x * blockDim.x;
  for (int i = tid; i < 20 * 64; i += stride) {
    int row = i >> 6, o = i & 63;
    float v;
    if (row < 16) {
      int r = row >> 2, ii = row & 3;
      v = 0.f;
      for (int b = 0; b < 4; ++b) v += comp1[r * 4 + b] * basis1[(b * 4 + ii) * 64 + o];
    } else {
      v = loop1[(row - 16) * 64 + o];
    }
    W1cat[i] = v;
  }
  for (int i = tid; i < 320 * 128; i += stride) {
    int row = i >> 7, o = i & 127;
    float v;
    if (row < 256) {
      int r = row >> 6, ii = row & 63;
      v = 0.f;
      for (int b = 0; b < 4; ++b) v += comp2[r * 4 + b] * basis2[(b * 64 + ii) * 128 + o];
    } else {
      v = loop2[(row - 256) * 128 + o];
    }
    W2cat[i] = v;
  }
}

// ---------------- K1: zero the scatter accumulators (every call) ----------------
__global__ void zero_kernel(float* __restrict__ p, int n) {
  int i = blockIdx.x * blockDim.x + threadIdx.x;
  int stride = gridDim.x * blockDim.x;
  for (; i < n; i += stride) p[i] = 0.f;
}

// ---------------- K2: layer-1 edge scatter (in-dim = 4) ----------------
__global__ void edge_agg1_kernel(const float* __restrict__ nf, const int* __restrict__ src,
                                 const int* __restrict__ dst, const int* __restrict__ et,
                                 float* __restrict__ s1) {
  int e = blockIdx.x * blockDim.x + threadIdx.x;
  if (e >= EE) return;
  int s = src[e], d = dst[e], t = et[e];
  const float* x = nf + s * 4;
  float* o = s1 + d * 16 + t * 4;
  atomicAdd(o + 0, x[0]);
  atomicAdd(o + 1, x[1]);
  atomicAdd(o + 2, x[2]);
  atomicAdd(o + 3, x[3]);
}

// ---------------- K3: layer-1 node transform (WMMA, K=20, out=64) + LN ----------------
__global__ __launch_bounds__(128) void node_l1_kernel(
    const float* __restrict__ s1, const float* __restrict__ nf,
    const float* __restrict__ W1cat, const float* __restrict__ rbias,
    const float* __restrict__ lng, const float* __restrict__ lnb,
    float* __restrict__ h1) {
  __shared__ float xcat[16 * 21];   // 16 nodes x 20 (padded)
  __shared__ float outb[16 * 65];   // 16 nodes x 64 (padded)
  int nb = blockIdx.x * 16, tid = threadIdx.x;
  for (int i = tid; i < 16 * 20; i += 128) {
    int n = i / 20, c = i % 20;
    float v = (c < 16) ? s1[(nb + n) * 16 + c] : nf[(nb + n) * 4 + (c - 16)];
    xcat[n * 21 + c] = v;
  }
  __syncthreads();
  int lane = tid & 31, wave = tid >> 5;
  int r = lane & 15;               // A row / C col-within-tile
  int khalf = (lane >> 4) * 2;     // K pair select
  int col = wave * 16 + r;         // output column (0..63)
  v8f c = {0.f, 0.f, 0.f, 0.f, 0.f, 0.f, 0.f, 0.f};
  for (int kk = 0; kk < 5; ++kk) {
    int k0 = kk * 4 + khalf;
    v2f a, b;
    a.x = xcat[r * 21 + k0];
    a.y = xcat[r * 21 + k0 + 1];
    b.x = W1cat[k0 * 64 + col];
    b.y = W1cat[(k0 + 1) * 64 + col];
    c = wmma4(a, b, c);
  }
  int m0 = (lane >> 4) * 8;
  float bias = rbias[col];
#pragma unroll
  for (int j = 0; j < 8; ++j) {
    int m = m0 + j;
    outb[m * 65 + col] = lrelu(c[j] + bias);
  }
  __syncthreads();
  if (tid < 16) {  // LayerNorm over 64 feats per node
    float mean = 0.f;
    for (int f = 0; f < 64; ++f) mean += outb[tid * 65 + f];
    mean *= (1.f / 64.f);
    float var = 0.f;
    for (int f = 0; f < 64; ++f) { float d = outb[tid * 65 + f] - mean; var += d * d; }
    var *= (1.f / 64.f);
    float rs = rsqrtf(var + 1e-5f);
    for (int f = 0; f < 64; ++f)
      h1[(nb + tid) * 64 + f] = lng[f] * (outb[tid * 65 + f] - mean) * rs + lnb[f];
  }
}

// ---------------- K4: layer-2 edge scatter (in-dim = 64) ----------------
__global__ void edge_agg2_kernel(const float* __restrict__ h1, const int* __restrict__ src,
                                 const int* __restrict__ dst, const int* __restrict__ et,
                                 float* __restrict__ s2) {
  int tid = blockIdx.x * blockDim.x + threadIdx.x;
  int e = tid >> 2;
  if (e >= EE) return;
  int q = tid & 3;  // 16-float chunk
  int s = src[e], d = dst[e], t = et[e];
  const float* x = h1 + s * 64 + q * 16;
  float* o = s2 + d * 256 + t * 64 + q * 16;
#pragma unroll
  for (int i = 0; i < 16; ++i) atomicAdd(o + i, x[i]);
}

// ---------------- K5: layer-2 node transform (WMMA, K=320, out=128) + LN ----------------
__global__ __launch_bounds__(128) void node_l2_kernel(
    const float* __restrict__ s2, const float* __restrict__ h1,
    const float* __restrict__ W2cat, const float* __restrict__ rbias,
    const float* __restrict__ lng, const float* __restrict__ lnb,
    float* __restrict__ hout) {
  __shared__ float xcat[16 * 321];  // 16 nodes x 320 (padded)
  __shared__ float outb[16 * 129];  // 16 nodes x 128 (padded)
  int nb = blockIdx.x * 16, tid = threadIdx.x;
  for (int i = tid; i < 16 * 320; i += 128) {
    int n = i / 320, c = i % 320;
    xcat[n * 321 + c] = (c < 256) ? s2[(nb + n) * 256 + c] : h1[(nb + n) * 64 + (c - 256)];
  }
  __syncthreads();
  int lane = tid & 31, wave = tid >> 5;
  int r = lane & 15, khalf = (lane >> 4) * 2;
  int col0 = wave * 32 + r, col1 = col0 + 16;
  v8f c0 = {0.f, 0.f, 0.f, 0.f, 0.f, 0.f, 0.f, 0.f};
  v8f c1 = {0.f, 0.f, 0.f, 0.f, 0.f, 0.f, 0.f, 0.f};
  for (int kk = 0; kk < 80; ++kk) {
    int k0 = kk * 4 + khalf;
    __builtin_prefetch(&W2cat[(kk * 4 + 16) * 128 + col0], 0, 1);
    v2f a, b0, b1;
    a.x = xcat[r * 321 + k0];
    a.y = xcat[r * 321 + k0 + 1];
    b0.x = W2cat[k0 * 128 + col0];
    b0.y = W2cat[(k0 + 1) * 128 + col0];
    b1.x = W2cat[k0 * 128 + col1];
    b1.y = W2cat[(k0 + 1) * 128 + col1];
    c0 = wmma4(a, b0, c0);
    c1 = wmma4(a, b1, c1);
  }
  int m0 = (lane >> 4) * 8;
  float bb0 = rbias[col0], bb1 = rbias[col1];
#pragma unroll
  for (int j = 0; j < 8; ++j) {
    int m = m0 + j;
    outb[m * 129 + col0] = lrelu(c0[j] + bb0);
    outb[m * 129 + col1] = lrelu(c1[j] + bb1);
  }
  __syncthreads();
  if (tid < 16) {  // LayerNorm over 128 feats
    float mean = 0.f;
    for (int f = 0; f < 128; ++f) mean += outb[tid * 129 + f];
    mean *= (1.f / 128.f);
    float var = 0.f;
    for (int f = 0; f < 128; ++f) { float d = outb[tid * 129 + f] - mean; var += d * d; }
    var *= (1.f / 128.f);
    float rs = rsqrtf(var + 1e-5f);
    for (int f = 0; f < 128; ++f)
      hout[(nb + tid) * 128 + f] = lng[f] * (outb[tid * 129 + f] - mean) * rs + lnb[f];
  }
}

// ---------------- K6: gate MLP (128 -> 64 -> 1), 8 nodes per block ----------------
__global__ __launch_bounds__(64) void gate_kernel(const float* __restrict__ h,
                                                  const float* __restrict__ gW1,
                                                  const float* __restrict__ gb1,
                                                  const float* __restrict__ gW2,
                                                  const float* __restrict__ gb2,
                                                  float* __restrict__ gate) {
  __shared__ float hsh[8 * 128];
  __shared__ float red[8 * 64];
  int nb = blockIdx.x * 8, tid = threadIdx.x;
  for (int i = tid; i < 8 * 128; i += 64) hsh[i] = h[nb * 128 + i];
  __syncthreads();
  float acc[8];
#pragma unroll
  for (int m = 0; m < 8; ++m) acc[m] = gb1[tid];
  for (int k = 0; k < 128; ++k) {
    float w = gW1[k * 64 + tid];
#pragma unroll
    for (int m = 0; m < 8; ++m) acc[m] += hsh[m * 128 + k] * w;
  }
  float w2 = gW2[tid];
#pragma unroll
  for (int m = 0; m < 8; ++m) red[m * 64 + tid] = relu_(acc[m]) * w2;
  __syncthreads();
  for (int s = 32; s > 0; s >>= 1) {
    if (tid < s) {
#pragma unroll
      for (int m = 0; m < 8; ++m) red[m * 64 + tid] += red[m * 64 + tid + s];
    }
    __syncthreads();
  }
  if (tid < 8) gate[nb + tid] = red[tid * 64] + gb2[0];
}

// ---------------- K7: per-graph pooling + VAE heads + edge recon ----------------
__global__ __launch_bounds__(128) void pool_vae_kernel(
    const float* __restrict__ h, const float* __restrict__ gate, const float* __restrict__ eps,
    const float* __restrict__ Wmu, const float* __restrict__ bmu,
    const float* __restrict__ Wlv, const float* __restrict__ blv,
    const float* __restrict__ Wlp, const float* __restrict__ blp,
    const float* __restrict__ We1, const float* __restrict__ be1,
    const float* __restrict__ We2, const float* __restrict__ be2,
    const float* __restrict__ Wen1, const float* __restrict__ ben1,
    const float* __restrict__ Wen2, const float* __restrict__ ben2,
    const float* __restrict__ Wst1, const float* __restrict__ bst1,
    const float* __restrict__ Wst2, const float* __restrict__ bst2,
    float* __restrict__ out_mu, float* __restrict__ out_lv, float* __restrict__ out_z,
    float* __restrict__ out_en, float* __restrict__ out_st, float* __restrict__ out_edge,
    float* __restrict__ zp_out) {
  __shared__ float att[128];
  __shared__ float red[128];
  __shared__ float gemb[128];
  __shared__ float zsh[32];
  __shared__ float zpsh[128];
  __shared__ float hidsh[64];
  __shared__ float e3[3];
  int b = blockIdx.x, tid = threadIdx.x;
  int nb = b * 128;
  float g = gate[nb + tid];
  red[tid] = g;
  __syncthreads();
  for (int s = 64; s > 0; s >>= 1) {
    if (tid < s) red[tid] = fmaxf(red[tid], red[tid + s]);
    __syncthreads();
  }
  float gmax = red[0];
  __syncthreads();
  float ex = expf(g - gmax);
  red[tid] = ex;
  __syncthreads();
  for (int s = 64; s > 0; s >>= 1) {
    if (tid < s) red[tid] += red[tid + s];
    __syncthreads();
  }
  float den = red[0];
  __syncthreads();
  att[tid] = ex / den;
  __syncthreads();
  // graph embedding: coalesced row reads of h
  float acc = 0.f;
  for (int n = 0; n < 128; ++n) acc += att[n] * h[(nb + n) * 128 + tid];
  gemb[tid] = acc;
  __syncthreads();
  // mu / logvar / z
  if (tid < 32) {
    float mu = bmu[tid], lv = blv[tid];
    for (int k = 0; k < 128; ++k) {
      float gk = gemb[k];
      mu += gk * Wmu[k * 32 + tid];
      lv += gk * Wlv[k * 32 + tid];
    }
    mu = fminf(fmaxf(mu, -5.f), 5.f);
    lv = fminf(fmaxf(lv, -10.f), 10.f);
    float sd = fminf(fmaxf(expf(0.5f * lv), 1e-6f), 1e6f);
    float z = mu + eps[b * 32 + tid] * sd;
    out_mu[b * 32 + tid] = mu;
    out_lv[b * 32 + tid] = lv;
    out_z[b * 32 + tid] = z;
    zsh[tid] = z;
  }
  __syncthreads();
  // zp = relu(z @ Wlp + blp)
  {
    float a = blp[tid];
    for (int k = 0; k < 32; ++k) a += zsh[k] * Wlp[k * 128 + tid];
    a = relu_(a);
    zpsh[tid] = a;
    zp_out[b * 128 + tid] = a;
  }
  __syncthreads();
  // energy head: zl = [z, 0_6] so only first 32 rows of Wen1 matter
  if (tid < 64) {
    float a = ben1[tid];
    for (int k = 0; k < 32; ++k) a += zsh[k] * Wen1[k * 64 + tid];
    hidsh[tid] = relu_(a);
  }
  __syncthreads();
  if (tid < 2) {
    float a = ben2[tid];
    for (int j = 0; j < 64; ++j) a += hidsh[j] * Wen2[j * 2 + tid];
    out_en[b * 2 + tid] = a;
  }
  __syncthreads();
  // stress head
  if (tid < 64) {
    float a = bst1[tid];
    for (int k = 0; k < 32; ++k) a += zsh[k] * Wst1[k * 64 + tid];
    hidsh[tid] = relu_(a);
  }
  __syncthreads();
  if (tid < 9) {
    float a = bst2[tid];
    for (int j = 0; j < 64; ++j) a += hidsh[j] * Wst2[j * 9 + tid];
    out_st[b * 9 + tid] = a;
  }
  __syncthreads();
  // edge recon: z_exp[src]==z_exp[dst]==zp[b] -> one MLP per graph, broadcast to 2048 edges
  if (tid < 64) {
    float a = be1[tid];
    for (int k = 0; k < 128; ++k)
      a += zpsh[k] * (We1[k * 64 + tid] + We1[(k + 128) * 64 + tid]);
    hidsh[tid] = relu_(a);
  }
  __syncthreads();
  if (tid < 3) {
    float a = be2[tid];
    for (int j = 0; j < 64; ++j) a += hidsh[j] * We2[j * 3 + tid];
    e3[tid] = a;
  }
  __syncthreads();
  float v0 = e3[0], v1 = e3[1], v2 = e3[2];
  for (int e = tid; e < EPER; e += 128) {
    int idx = (b * EPER + e) * 3;
    out_edge[idx] = v0;
    out_edge[idx + 1] = v1;
    out_edge[idx + 2] = v2;
  }
}

// ---------------- K8: recon_node via chained WMMA (128->128, 128->64, 64->4) ----------------
__global__ __launch_bounds__(128) void recon_node_kernel(
    const float* __restrict__ h, const float* __restrict__ zp,
    const float* __restrict__ Wnp, const float* __restrict__ bnp,
    const float* __restrict__ Wn1, const float* __restrict__ bn1,
    const float* __restrict__ Wn2, const float* __restrict__ bn2,
    float* __restrict__ recon) {
  __shared__ float hs[16 * 129];
  __shared__ float ts[16 * 129];
  __shared__ float hid[16 * 65];
  int nb = blockIdx.x * 16, tid = threadIdx.x;
  int b = nb >> 7;  // graph id (16-node tiles never straddle 128-node graphs)
  for (int i = tid; i < 16 * 128; i += 128) {
    int n = i >> 7, c = i & 127;
    hs[n * 129 + c] = h[(nb + n) * 128 + c];
  }
  __syncthreads();
  int lane = tid & 31, wave = tid >> 5;
  int r = lane & 15, khalf = (lane >> 4) * 2;
  int m0 = (lane >> 4) * 8;
  // stage 1: nep = relu(h @ Wnp + bnp); t = zp[b] + nep
  {
    v8f c0 = {0.f, 0.f, 0.f, 0.f, 0.f, 0.f, 0.f, 0.f};
    v8f c1 = {0.f, 0.f, 0.f, 0.f, 0.f, 0.f, 0.f, 0.f};
    int col0 = wave * 32 + r, col1 = col0 + 16;
    for (int kk = 0; kk < 32; ++kk) {
      int k0 = kk * 4 + khalf;
      v2f a, b0, b1;
      a.x = hs[r * 129 + k0];
      a.y = hs[r * 129 + k0 + 1];
      b0.x = Wnp[k0 * 128 + col0];
      b0.y = Wnp[(k0 + 1) * 128 + col0];
      b1.x = Wnp[k0 * 128 + col1];
      b1.y = Wnp[(k0 + 1) * 128 + col1];
      c0 = wmma4(a, b0, c0);
      c1 = wmma4(a, b1, c1);
    }
    float z0 = zp[b * 128 + col0], z1 = zp[b * 128 + col1];
    float bb0 = bnp[col0], bb1 = bnp[col1];
#pragma unroll
    for (int j = 0; j < 8; ++j) {
      int m = m0 + j;
      ts[m * 129 + col0] = z0 + relu_(c0[j] + bb0);
      ts[m * 129 + col1] = z1 + relu_(c1[j] + bb1);
    }
  }
  __syncthreads();
  // stage 2: hidden = relu(t @ Wn1 + bn1), 64 cols -> one 16-col tile per wave
  {
    v8f c0 = {0.f, 0.f, 0.f, 0.f, 0.f, 0.f, 0.f, 0.f};
    int col0 = wave * 16 + r;
    for (int kk = 0; kk < 32; ++kk) {
      int k0 = kk * 4 + khalf;
      v2f a, b0;
      a.x = ts[r * 129 + k0];
      a.y = ts[r * 129 + k0 + 1];
      b0.x = Wn1[k0 * 64 + col0];
      b0.y = Wn1[(k0 + 1) * 64 + col0];
      c0 = wmma4(a, b0, c0);
    }
    float bb0 = bn1[col0];
#pragma unroll
    for (int j = 0; j < 8; ++j) {
      int m = m0 + j;
      hid[m * 65 + col0] = relu_(c0[j] + bb0);
    }
  }
  __syncthreads();
  // stage 3: recon = hidden @ Wn2 + bn2 (64 -> 4), tiny
  if (tid < 64) {
    int n = tid >> 2, c = tid & 3;
    float a = bn2[c];
    for (int j = 0; j < 64; ++j) a += hid[n * 65 + j] * Wn2[j * 4 + c];
    recon[(nb + n) * 4 + c] = a;
  }
}

// ---------------- host launcher ----------------
extern "C" void kernel_launch(void* const* d_in, const int* in_sizes, int n_in,
                              void* d_out, int out_size, void* d_ws, size_t ws_size,
                              hipStream_t stream) {
  (void)in_sizes; (void)n_in; (void)out_size; (void)ws_size;
  // inputs (setup_inputs order, params flattened in dict order)
  const float* node_feats = (const float*)d_in[0];
  const int*   src        = (const int*)d_in[1];
  const int*   dst        = (const int*)d_in[2];
  const int*   etypes     = (const int*)d_in[3];
  const float* eps        = (const float*)d_in[4];
  const float* basis1 = (const float*)d_in[5];
  const float* comp1  = (const float*)d_in[6];
  const float* loop1  = (const float*)d_in[7];
  const float* rbias1 = (const float*)d_in[8];
  const float* ln1g   = (const float*)d_in[9];
  const float* ln1b   = (const float*)d_in[10];
  const float* basis2 = (const float*)d_in[11];
  const float* comp2  = (const float*)d_in[12];
  const float* loop2  = (const float*)d_in[13];
  const float* rbias2 = (const float*)d_in[14];
  const float* ln2g   = (const float*)d_in[15];
  const float* ln2b   = (const float*)d_in[16];
  const float* gW1 = (const float*)d_in[17];
  const float* gb1 = (const float*)d_in[18];
  const float* gW2 = (const float*)d_in[19];
  const float* gb2 = (const float*)d_in[20];
  const float* Wmu = (const float*)d_in[21];
  const float* bmu = (const float*)d_in[22];
  const float* Wlv = (const float*)d_in[23];
  const float* blv = (const float*)d_in[24];
  const float* Wlp = (const float*)d_in[25];
  const float* blp = (const float*)d_in[26];
  const float* Wnp = (const float*)d_in[27];
  const float* bnp = (const float*)d_in[28];
  const float* Wn1 = (const float*)d_in[29];
  const float* bn1 = (const float*)d_in[30];
  const float* Wn2 = (const float*)d_in[31];
  const float* bn2 = (const float*)d_in[32];
  const float* We1 = (const float*)d_in[33];
  const float* be1 = (const float*)d_in[34];
  const float* We2 = (const float*)d_in[35];
  const float* be2 = (const float*)d_in[36];
  const float* Wen1 = (const float*)d_in[37];
  const float* ben1 = (const float*)d_in[38];
  const float* Wen2 = (const float*)d_in[39];
  const float* ben2 = (const float*)d_in[40];
  const float* Wst1 = (const float*)d_in[41];
  const float* bst1 = (const float*)d_in[42];
  const float* Wst2 = (const float*)d_in[43];
  const float* bst2 = (const float*)d_in[44];

  // output layout (dict order): mu, logvar, z, node_emb, recon_node, recon_edge, energy, stress
  float* out = (float*)d_out;
  float* o_mu    = out;
  float* o_lv    = out + NB * LAT;
  float* o_z     = o_lv + NB * LAT;
  float* o_h     = o_z + NB * LAT;                 // N * 128
  float* o_rnode = o_h + (size_t)NN * 128;         // N * 4
  float* o_redge = o_rnode + (size_t)NN * 4;       // E * 3
  float* o_en    = o_redge + (size_t)EE * 3;       // B * 2
  float* o_st    = o_en + NB * 2;                  // B * 9

  // workspace layout
  float* ws   = (float*)d_ws;
  float* s1   = ws;                                // N*16
  float* s2   = s1 + (size_t)NN * 16;              // N*256
  float* h1w  = s2 + (size_t)NN * 256;             // N*64
  float* W1c  = h1w + (size_t)NN * 64;             // 20*64
  float* W2c  = W1c + 20 * 64;                     // 320*128
  float* gatw = W2c + 320 * 128;                   // N
  float* zpw  = gatw + NN;                         // B*128

  const int zero_n = NN * 16 + NN * 256;           // s1 + s2 accumulators

  prep_w_kernel<<<96, 256, 0, stream>>>(basis1, comp1, loop1, basis2, comp2, loop2, W1c, W2c);
  zero_kernel<<<2048, 256, 0, stream>>>(s1, zero_n);
  edge_agg1_kernel<<<EE / 256, 256, 0, stream>>>(node_feats, src, dst, etypes, s1);
  node_l1_kernel<<<NN / 16, 128, 0, stream>>>(s1, node_feats, W1c, rbias1, ln1g, ln1b, h1w);
  edge_agg2_kernel<<<(EE * 4) / 256, 256, 0, stream>>>(h1w, src, dst, etypes, s2);
  node_l2_kernel<<<NN / 16, 128, 0, stream>>>(s2, h1w, W2c, rbias2, ln2g, ln2b, o_h);
  gate_kernel<<<NN / 8, 64, 0, stream>>>(o_h, gW1, gb1, gW2, gb2, gatw);
  pool_vae_kernel<<<NB, 128, 0, stream>>>(o_h, gatw, eps, Wmu, bmu, Wlv, blv, Wlp, blp,
                                          We1, be1, We2, be2, Wen1, ben1, Wen2, ben2,
                                          Wst1, bst1, Wst2, bst2,
                                          o_mu, o_lv, o_z, o_en, o_st, o_redge, zpw);
  recon_node_kernel<<<NN / 16, 128, 0, stream>>>(o_h, zpw, Wnp, bnp, Wn1, bn1, Wn2, bn2, o_rnode);
}